// BERT_CRF_51539608032
// MI455X (gfx1250) — compile-verified
//
#include <hip/hip_runtime.h>
#include <hip/hip_bf16.h>
#include <math.h>

typedef __attribute__((ext_vector_type(2))) float v2f;
typedef __attribute__((ext_vector_type(8))) float v8f;

#define BB 64
#define SS 512
#define DD 768
#define TT 9
#define TP 16    // tag dim padded to 16 for WMMA N
#define KP 770   // LDS k-stride in floats (avoids 64-bank conflicts on b64 reads)

// ---------------------------------------------------------------------------
// Kernel 1: emissions[b,s,t] = embeds[b,s,:] . weight[t,:] + bias[t]
// FP32 WMMA 16x16x4. One wave per 16-row tile of the (B*S) dimension.
// ---------------------------------------------------------------------------
__global__ __launch_bounds__(256)
void emissions_wmma_kernel(const float* __restrict__ embeds,
                           const float* __restrict__ weight,
                           const float* __restrict__ bias,
                           float* __restrict__ emis) {
    __shared__ float wlds[16 * KP];

    const int tid = threadIdx.x;
    // Stage zero-padded weight (16 x 768, row-major [n][k]) into LDS.
    for (int idx = tid; idx < 16 * DD; idx += 256) {
        const int n = idx / DD;
        const int k = idx - n * DD;
        wlds[n * KP + k] = (n < TT) ? weight[n * DD + k] : 0.0f;
    }
    __syncthreads();

    const int wave = tid >> 5;
    const int lane = tid & 31;
    const int half = lane >> 4;   // selects K pair within the fragment
    const int ln   = lane & 15;   // M index for A, N index for B
    const int rowbase = (blockIdx.x * 8 + wave) * 16;

    const float* __restrict__ arow = embeds + (size_t)(rowbase + ln) * DD + 2 * half;
    const float* __restrict__ brow = &wlds[ln * KP + 2 * half];

    v8f c = {};
    #pragma unroll 4
    for (int k0 = 0; k0 < DD; k0 += 4) {
        // A fragment: lane&15 = M row; half picks K={k0,k0+1} or {k0+2,k0+3}
        v2f a = *(const v2f*)(arow + k0);
        // B fragment: lane&15 = N col; same half/K mapping
        v2f b = *(const v2f*)(brow + k0);
        // 8 args: (neg_a, A, neg_b, B, c_mod, C, reuse_a, reuse_b)
        c = __builtin_amdgcn_wmma_f32_16x16x4_f32(false, a, false, b,
                                                  (short)0, c, false, false);
    }

    // C layout: VGPR r -> M = r + 8*half, N = lane&15
    const float bv = (ln < TT) ? bias[ln] : 0.0f;
    #pragma unroll
    for (int r = 0; r < 8; ++r) {
        const int row = r + 8 * half;
        emis[(size_t)(rowbase + row) * TP + ln] = c[r] + bv;
    }
}

// ---------------------------------------------------------------------------
// Kernel 2: CRF forward scan (denominator). One wave per batch; lane = tag.
// ---------------------------------------------------------------------------
__global__ __launch_bounds__(32)
void crf_scan_kernel(const float* __restrict__ emis,
                     const float* __restrict__ transitions,
                     const float* __restrict__ start_t,
                     const float* __restrict__ end_t,
                     const unsigned char* __restrict__ mask,
                     float* __restrict__ den) {
    const int b    = blockIdx.x;
    const int lane = threadIdx.x;
    const int tp   = (lane < TT) ? lane : (TT - 1);   // clamp inactive lanes

    float trow[TT];
    #pragma unroll
    for (int t = 0; t < TT; ++t) trow[t] = transitions[t * TT + tp];

    const float* __restrict__ eb = emis + (size_t)b * SS * TP;
    const unsigned char* __restrict__ mb = mask + (size_t)b * SS;

    float score = start_t[tp] + eb[tp];

    for (int s = 1; s < SS; ++s) {
        const float e = eb[(size_t)s * TP + tp];
        float v[TT];
        float m = -3.0e38f;
        #pragma unroll
        for (int t = 0; t < TT; ++t) {
            const float st = __shfl(score, t, 32);   // broadcast score[t]
            v[t] = st + trow[t];
            m = fmaxf(m, v[t]);
        }
        float sum = 0.0f;
        #pragma unroll
        for (int t = 0; t < TT; ++t) sum += __expf(v[t] - m);
        const float nxt = m + __logf(sum) + e;
        score = (mb[s] != 0) ? nxt : score;
    }

    // den[b] = logsumexp_t(score[t] + end[t])
    float val = (lane < TT) ? (score + end_t[lane]) : -INFINITY;
    float mm = val;
    #pragma unroll
    for (int off = 16; off >= 1; off >>= 1) mm = fmaxf(mm, __shfl_xor(mm, off, 32));
    float se = __expf(val - mm);
    #pragma unroll
    for (int off = 16; off >= 1; off >>= 1) se += __shfl_xor(se, off, 32);
    if (lane == 0) den[b] = mm + __logf(se);
}

// ---------------------------------------------------------------------------
// Kernel 3: numerator. One wave per batch; lanes stride over sequence.
// ---------------------------------------------------------------------------
__global__ __launch_bounds__(32)
void crf_num_kernel(const float* __restrict__ emis,
                    const float* __restrict__ transitions,
                    const float* __restrict__ start_t,
                    const float* __restrict__ end_t,
                    const int* __restrict__ tags,
                    const unsigned char* __restrict__ mask,
                    float* __restrict__ num) {
    const int b    = blockIdx.x;
    const int lane = threadIdx.x;
    const int* __restrict__ tb = tags + (size_t)b * SS;
    const unsigned char* __restrict__ mb = mask + (size_t)b * SS;
    const float* __restrict__ eb = emis + (size_t)b * SS * TP;

    float acc = 0.0f;
    int   cnt = 0;
    for (int s = lane; s < SS; s += 32) {
        const int tg = tb[s];
        const int mk = (mb[s] != 0) ? 1 : 0;
        cnt += mk;
        const float e = eb[(size_t)s * TP + tg];
        if (s == 0) {
            acc += start_t[tg] + e;                       // unmasked first step
        } else if (mk) {
            acc += transitions[tb[s - 1] * TT + tg] + e;  // masked transitions
        }
    }
    #pragma unroll
    for (int off = 16; off >= 1; off >>= 1) {
        acc += __shfl_xor(acc, off, 32);
        cnt += __shfl_xor(cnt, off, 32);
    }
    if (lane == 0) {
        const int seq_end = cnt - 1;
        num[b] = acc + end_t[tb[seq_end]];
    }
}

// ---------------------------------------------------------------------------
// Kernel 4: loss = -mean(num - den)
// ---------------------------------------------------------------------------
__global__ __launch_bounds__(32)
void finalize_kernel(const float* __restrict__ num,
                     const float* __restrict__ den,
                     float* __restrict__ out) {
    const int lane = threadIdx.x;
    float v = 0.0f;
    for (int i = lane; i < BB; i += 32) v += num[i] - den[i];
    #pragma unroll
    for (int off = 16; off >= 1; off >>= 1) v += __shfl_xor(v, off, 32);
    if (lane == 0) out[0] = -v / (float)BB;
}

// ---------------------------------------------------------------------------
// Launch
// ---------------------------------------------------------------------------
extern "C" void kernel_launch(void* const* d_in, const int* in_sizes, int n_in,
                              void* d_out, int out_size, void* d_ws, size_t ws_size,
                              hipStream_t stream) {
    const float*         embeds  = (const float*)d_in[0];
    const float*         weight  = (const float*)d_in[1];
    const float*         bias    = (const float*)d_in[2];
    const float*         start_t = (const float*)d_in[3];
    const float*         end_t   = (const float*)d_in[4];
    const float*         trans   = (const float*)d_in[5];
    const int*           tags    = (const int*)d_in[6];
    const unsigned char* mask    = (const unsigned char*)d_in[7];

    float* emis = (float*)d_ws;                        // B*S*16 floats (2 MB)
    float* den  = emis + (size_t)BB * SS * TP;         // 64 floats
    float* num  = den + BB;                            // 64 floats

    // 32768 rows / 16 per wave / 8 waves per block = 256 blocks
    emissions_wmma_kernel<<<256, 256, 0, stream>>>(embeds, weight, bias, emis);
    crf_scan_kernel<<<BB, 32, 0, stream>>>(emis, trans, start_t, end_t, mask, den);
    crf_num_kernel<<<BB, 32, 0, stream>>>(emis, trans, start_t, end_t, tags, mask, num);
    finalize_kernel<<<1, 32, 0, stream>>>(num, den, (float*)d_out);
}